// InvGridSamplerNumerator_3066606649873
// MI455X (gfx1250) — compile-verified
//
#include <hip/hip_runtime.h>
#include <stdint.h>

// Problem shape (fixed by the reference): x:(8,16,512,512) f32, inv_grid:(8,512,512,2) f32
// out:(8,16,512,512) f32. Inverse bilinear scatter-add with weights shared across C.
namespace {
constexpr int B = 8, C = 16, H = 512, W = 512;
constexpr int HW = H * W;
constexpr int THREADS = 256;
}

// Typed pointers for the async global->LDS builtin (b64 moves one int2 per lane).
typedef int v2i __attribute__((ext_vector_type(2)));
typedef __attribute__((address_space(1))) v2i* v2i_gptr;
typedef __attribute__((address_space(3))) v2i* v2i_lptr;

__global__ __launch_bounds__(256) void zero_f4(float4* __restrict__ o, int n4) {
  int i = blockIdx.x * blockDim.x + threadIdx.x;
  if (i < n4) o[i] = float4{0.f, 0.f, 0.f, 0.f};
}

__global__ __launch_bounds__(256) void zero_tail(float* __restrict__ o, int lo, int n) {
  int i = lo + threadIdx.x;
  if (i < n) o[i] = 0.f;
}

// Device-scope relaxed FP32 add, result discarded -> non-returning global_atomic_add_f32.
__device__ __forceinline__ void atomAddF32(float* p, float v) {
  __hip_atomic_fetch_add(p, v, __ATOMIC_RELAXED, __HIP_MEMORY_SCOPE_AGENT);
}

__global__ __launch_bounds__(256) void splat(const float* __restrict__ x,
                                             const float* __restrict__ grid,
                                             float* __restrict__ out,
                                             int npix) {
  __shared__ float sgrid[2 * THREADS];
  const int tid = threadIdx.x;
  const int pix = blockIdx.x * THREADS + tid;
  if (pix >= npix) return;

  // ---- Stage this pixel's (g0,g1) pair through LDS via the CDNA5 async path ----
  const float* gsrc = grid + 2 * (size_t)pix;
  float g0, g1;  // g0 -> row coordinate (gi), g1 -> col coordinate (gj)
#if __has_builtin(__builtin_amdgcn_global_load_async_to_lds_b64)
  __builtin_amdgcn_global_load_async_to_lds_b64((v2i_gptr)gsrc,
                                                (v2i_lptr)(&sgrid[2 * tid]),
                                                /*offset=*/0, /*cpol=*/0);
  asm volatile("s_wait_asynccnt 0" ::: "memory");  // ASYNCcnt==0: LDS data valid
  g0 = sgrid[2 * tid + 0];
  g1 = sgrid[2 * tid + 1];
#else
  g0 = gsrc[0];
  g1 = gsrc[1];
#endif

  const int b  = pix / HW;
  const int ij = pix - b * HW;  // i*W + j within the image plane

  // Reference math: g=(inv+1)*0.5; gi=clip(g0*H+1, 0, H+1-2e-10). In f32 the upper
  // bound 513-2e-10 rounds to exactly 513.0f.
  const float t0 = (g0 + 1.0f) * 0.5f;
  const float t1 = (g1 + 1.0f) * 0.5f;
  float gi = fminf(fmaxf(t0 * (float)H + 1.0f, 0.0f), (float)(H + 1));
  float gj = fminf(fmaxf(t1 * (float)W + 1.0f, 0.0f), (float)(W + 1));

  const int fi = (int)gi;  // gi >= 0, trunc == floor
  const int fj = (int)gj;
  const float fri = gi - (float)fi;  // exact (Sterbenz)
  const float frj = gj - (float)fj;
  const float wi0 = 1.0f - fri, wi1 = fri;
  const float wj0 = 1.0f - frj, wj1 = frj;

  // A-row = fi+di; after the [1:H+1] crop, output row = fi+di-1, valid iff in [0,H).
  const int r0 = fi - 1, r1 = fi;
  const int c0 = fj - 1, c1 = fj;
  const bool vr0 = (unsigned)r0 < (unsigned)H;
  const bool vr1 = (unsigned)r1 < (unsigned)H;
  const bool vc0 = (unsigned)c0 < (unsigned)W;
  const bool vc1 = (unsigned)c1 < (unsigned)W;

  // Issue all 16 channel loads up front (coalesced across adjacent j per plane;
  // 16 outstanding loads per thread hide L2/HBM latency behind the atomic burst).
  const float* xp = x + (size_t)b * C * HW + (size_t)ij;
  float v[C];
#pragma unroll
  for (int c = 0; c < C; ++c) v[c] = xp[(size_t)c * HW];

  float* ob = out + (size_t)b * C * HW;

  // Corner-outer / channel-inner: only 4 EXEC-predication points total (instead of
  // 64); the interior-pixel common path is 4 straight-line bursts of 16 atomics.
  {
    const float wgt = wi0 * wj0;
    if (vr0 && vc0) {
      float* p = ob + (r0 * W + c0);
#pragma unroll
      for (int c = 0; c < C; ++c) atomAddF32(p + (size_t)c * HW, v[c] * wgt);
    }
  }
  {
    const float wgt = wi0 * wj1;
    if (vr0 && vc1) {
      float* p = ob + (r0 * W + c1);
#pragma unroll
      for (int c = 0; c < C; ++c) atomAddF32(p + (size_t)c * HW, v[c] * wgt);
    }
  }
  {
    const float wgt = wi1 * wj0;
    if (vr1 && vc0) {
      float* p = ob + (r1 * W + c0);
#pragma unroll
      for (int c = 0; c < C; ++c) atomAddF32(p + (size_t)c * HW, v[c] * wgt);
    }
  }
  {
    const float wgt = wi1 * wj1;
    if (vr1 && vc1) {
      float* p = ob + (r1 * W + c1);
#pragma unroll
      for (int c = 0; c < C; ++c) atomAddF32(p + (size_t)c * HW, v[c] * wgt);
    }
  }
}

extern "C" void kernel_launch(void* const* d_in, const int* in_sizes, int n_in,
                              void* d_out, int out_size, void* d_ws, size_t ws_size,
                              hipStream_t stream) {
  (void)n_in; (void)d_ws; (void)ws_size;
  const float* x    = (const float*)d_in[0];
  const float* grid = (const float*)d_in[1];
  float* out = (float*)d_out;

  // Zero the accumulator (atomics below require a clean slate every call).
  const int n4 = out_size / 4;
  if (n4 > 0) zero_f4<<<(n4 + 255) / 256, 256, 0, stream>>>((float4*)out, n4);
  if (out_size - n4 * 4 > 0)
    zero_tail<<<1, 256, 0, stream>>>(out, n4 * 4, out_size);

  const int npix = in_sizes[1] / 2;  // B*H*W
  const int blocks = (npix + THREADS - 1) / THREADS;
  splat<<<blocks, THREADS, 0, stream>>>(x, grid, out, npix);
}